// CADGroupingGNN_70248485093418
// MI455X (gfx1250) — compile-verified
//
#include <hip/hip_runtime.h>
#include <hip/hip_bf16.h>
#include <cmath>

typedef __attribute__((ext_vector_type(2))) float v2f;
typedef __attribute__((ext_vector_type(8))) float v8f;

#define KDIM 128   // inner dim of every GEMM and node feature stride

// ---------------------------------------------------------------------------
// Small elementwise / edge kernels
// ---------------------------------------------------------------------------
__global__ __launch_bounds__(256) void k_init_deg(float* deg, int n) {
    int i = blockIdx.x * blockDim.x + threadIdx.x;
    if (i < n) deg[i] = 1.0f;                       // self-loop weight
}

__global__ __launch_bounds__(256) void k_ew_deg(const float* __restrict__ emb,
                                                const int* __restrict__ eattr,
                                                const int* __restrict__ ei,
                                                float* __restrict__ ew,
                                                float* __restrict__ deg, int E) {
    int e = blockIdx.x * blockDim.x + threadIdx.x;
    if (e >= E) return;
    float w = emb[eattr[e]];                        // Embedding(edge_attr)
    ew[e] = w;
    atomicAdd(&deg[ei[E + e]], w);                  // col = target node
}

__global__ __launch_bounds__(256) void k_dinv(const float* __restrict__ deg,
                                              float* __restrict__ dinv, int n) {
    int i = blockIdx.x * blockDim.x + threadIdx.x;
    if (i < n) {
        float d = deg[i];
        dinv[i] = d > 0.0f ? rsqrtf(d) : 0.0f;
    }
}

__global__ __launch_bounds__(256) void k_norm(const int* __restrict__ ei,
                                              const float* __restrict__ ew,
                                              const float* __restrict__ dinv,
                                              float* __restrict__ nrm, int E) {
    int e = blockIdx.x * blockDim.x + threadIdx.x;
    if (e >= E) return;
    nrm[e] = dinv[ei[e]] * ew[e] * dinv[ei[E + e]];
}

// out = bias + dinv^2 * h   (self-loop contribution + bias init)
__global__ __launch_bounds__(256) void k_agg_init(float* __restrict__ out,
                                                  const float* __restrict__ h,
                                                  const float* __restrict__ bias,
                                                  const float* __restrict__ dinv,
                                                  int total) {
    int idx = blockIdx.x * blockDim.x + threadIdx.x;
    if (idx >= total) return;
    int i = idx >> 7;                               // node
    int f = idx & 127;                              // feature
    float di = dinv[i];
    out[idx] = bias[f] + di * di * h[idx];
}

// one wave (32 lanes) per edge; 4 floats/lane; L2-resident fp32 atomics
__global__ __launch_bounds__(256) void k_agg_edges(float* __restrict__ out,
                                                   const float* __restrict__ h,
                                                   const float* __restrict__ nrm,
                                                   const int* __restrict__ ei, int E) {
    int t = blockIdx.x * blockDim.x + threadIdx.x;
    int e = t >> 5;
    if (e >= E) return;
    int lane = t & 31;
    int r = ei[e];                                  // source
    int c = ei[E + e];                              // target
    float w = nrm[e];
    const float4 hv = *(const float4*)(h + (size_t)r * KDIM + lane * 4);
    float* op = out + (size_t)c * KDIM + lane * 4;
    atomicAdd(op + 0, w * hv.x);
    atomicAdd(op + 1, w * hv.y);
    atomicAdd(op + 2, w * hv.z);
    atomicAdd(op + 3, w * hv.w);
}

__global__ __launch_bounds__(256) void k_relu(float* x, int n) {
    int i = blockIdx.x * blockDim.x + threadIdx.x;
    if (i < n) x[i] = fmaxf(x[i], 0.0f);
}

// ---------------------------------------------------------------------------
// WMMA FP32 GEMM: out[M x NC] = X[M x 128] @ W[128 x NC] (+ epilogue)
// One wave per 16-row strip. NTILES = NC/16 accumulator tiles held in VGPRs,
// A fragment reused across column tiles. Uses V_WMMA_F32_16X16X4_F32.
// EPI: 0 = none, 1 = bias+relu, 2 = bias+sigmoid
// ---------------------------------------------------------------------------
template <int NTILES, int EPI>
__global__ __launch_bounds__(32)
void gemm_wmma_f32(const float* __restrict__ X, const float* __restrict__ W,
                   const float* __restrict__ bias, float* __restrict__ out) {
    constexpr int NC = NTILES * 16;
    const int lane = threadIdx.x;
    const int half = lane >> 4;          // 0: lanes 0-15, 1: lanes 16-31
    const int l16  = lane & 15;
    const size_t row0 = (size_t)blockIdx.x * 16;
    const float* xrow = X + (row0 + l16) * KDIM;

    v8f acc[NTILES];
#pragma unroll
    for (int nt = 0; nt < NTILES; ++nt) acc[nt] = (v8f)0.0f;

#pragma unroll 2
    for (int kt = 0; kt < KDIM / 4; ++kt) {
        const int k0 = kt * 4 + 2 * half;          // ISA 16x4 f32 A layout
        const float2 av = *(const float2*)(xrow + k0);
        v2f a; a.x = av.x; a.y = av.y;
#pragma unroll
        for (int nt = 0; nt < NTILES; ++nt) {
            const float* wp = W + (size_t)k0 * NC + nt * 16 + l16;
            v2f b; b.x = wp[0]; b.y = wp[NC];      // B rows K=k0, k0+1
            acc[nt] = __builtin_amdgcn_wmma_f32_16x16x4_f32(
                false, a, false, b, (short)0, acc[nt], false, false);
        }
    }

#pragma unroll
    for (int nt = 0; nt < NTILES; ++nt) {
        const int col = nt * 16 + l16;
#pragma unroll
        for (int r = 0; r < 8; ++r) {
            float v = acc[nt][r];                  // C/D: M = r + 8*half
            if (EPI != 0) {                        // compile-time: no +0.0 adds
                v += bias[col];
                if (EPI == 1) v = fmaxf(v, 0.0f);
                if (EPI == 2) v = 1.0f / (1.0f + __expf(-v));
            }
            out[(row0 + r + 8 * half) * NC + col] = v;
        }
    }
}

// ---------------------------------------------------------------------------
extern "C" void kernel_launch(void* const* d_in, const int* in_sizes, int n_in,
                              void* d_out, int out_size, void* d_ws, size_t ws_size,
                              hipStream_t stream) {
    const float* x     = (const float*)d_in[0];
    const float* emb   = (const float*)d_in[1];
    const float* W1    = (const float*)d_in[2];
    const float* b1    = (const float*)d_in[3];
    const float* W2    = (const float*)d_in[4];
    const float* b2    = (const float*)d_in[5];
    const float* W3    = (const float*)d_in[6];
    const float* b3    = (const float*)d_in[7];
    const float* cW1   = (const float*)d_in[8];
    const float* cb1   = (const float*)d_in[9];
    const float* cW2   = (const float*)d_in[10];
    const float* cb2   = (const float*)d_in[11];
    const int*   ei    = (const int*)d_in[12];
    const int*   eattr = (const int*)d_in[13];

    const int N = in_sizes[0] / KDIM;     // 50000
    const int E = in_sizes[13];           // 800000

    // workspace layout (floats)
    float* ws   = (float*)d_ws;
    float* ew   = ws;                       // E
    float* nrm  = ew + E;                   // E
    float* deg  = nrm + E;                  // N
    float* dinv = deg + N;                  // N
    float* bufA = dinv + N;                 // N*128  (gemm output h)
    float* bufB = bufA + (size_t)N * KDIM;  // N*128  (aggregated output)

    const int TPB = 256;
    const int gN    = (N + TPB - 1) / TPB;
    const int gE    = (E + TPB - 1) / TPB;
    const int gNF   = (N * KDIM + TPB - 1) / TPB;
    const int gEdge = (E * 32 + TPB - 1) / TPB;
    const int gRows = N / 16;               // 50000/16 = 3125 exact

    // --- shared normalization (same row/col/w for all 3 layers) ---
    k_init_deg<<<gN, TPB, 0, stream>>>(deg, N);
    k_ew_deg<<<gE, TPB, 0, stream>>>(emb, eattr, ei, ew, deg, E);
    k_dinv<<<gN, TPB, 0, stream>>>(deg, dinv, N);
    k_norm<<<gE, TPB, 0, stream>>>(ei, ew, dinv, nrm, E);

    // --- GCN layer 1 ---
    gemm_wmma_f32<8, 0><<<gRows, 32, 0, stream>>>(x, W1, nullptr, bufA);
    k_agg_init<<<gNF, TPB, 0, stream>>>(bufB, bufA, b1, dinv, N * KDIM);
    k_agg_edges<<<gEdge, TPB, 0, stream>>>(bufB, bufA, nrm, ei, E);
    k_relu<<<gNF, TPB, 0, stream>>>(bufB, N * KDIM);

    // --- GCN layer 2 ---
    gemm_wmma_f32<8, 0><<<gRows, 32, 0, stream>>>(bufB, W2, nullptr, bufA);
    k_agg_init<<<gNF, TPB, 0, stream>>>(bufB, bufA, b2, dinv, N * KDIM);
    k_agg_edges<<<gEdge, TPB, 0, stream>>>(bufB, bufA, nrm, ei, E);
    k_relu<<<gNF, TPB, 0, stream>>>(bufB, N * KDIM);

    // --- GCN layer 3 ---
    gemm_wmma_f32<8, 0><<<gRows, 32, 0, stream>>>(bufB, W3, nullptr, bufA);
    k_agg_init<<<gNF, TPB, 0, stream>>>(bufB, bufA, b3, dinv, N * KDIM);
    k_agg_edges<<<gEdge, TPB, 0, stream>>>(bufB, bufA, nrm, ei, E);
    k_relu<<<gNF, TPB, 0, stream>>>(bufB, N * KDIM);

    // --- classifier head ---
    gemm_wmma_f32<8, 1><<<gRows, 32, 0, stream>>>(bufB, cW1, cb1, bufA); // relu
    gemm_wmma_f32<4, 2><<<gRows, 32, 0, stream>>>(bufA, cW2, cb2,
                                                  (float*)d_out);        // sigmoid
}